// CostVolumeBuilder_38809324487104
// MI455X (gfx1250) — compile-verified
//
#include <hip/hip_runtime.h>

typedef __attribute__((ext_vector_type(2))) float v2f;
typedef __attribute__((ext_vector_type(8))) float v8f;

#define BATCH   8
#define CH      128
#define HH      128
#define WW      256
#define NDISP   48
#define HWPLANE (HH * WW)   // 32768 elements per channel plane
#define WTILES  (WW / 16)   // 16 w-tiles per row

// K-loop over C=128 channels, 4 per V_WMMA_F32_16X16X4_F32.
// MASKED=true only for left-edge tiles (w0 < 48) that need zero-padding.
template <bool MASKED>
__device__ __forceinline__ void cv_kloop(
    const float* __restrict__ lp,
    const float* __restrict__ rp0, const float* __restrict__ rp1,
    const float* __restrict__ rp2, const float* __restrict__ rp3,
    float bm0, float bm1, float bm2, float bm3,
    int hf, v8f& acc0, v8f& acc1, v8f& acc2, v8f& acc3)
{
    #pragma unroll 4
    for (int kb = 0; kb < CH / 4; ++kb) {
        const size_t o0 = (size_t)(kb * 4 + hf * 2) * HWPLANE;
        const size_t o1 = o0 + HWPLANE;

        v2f a;
        a.x = lp[o0];
        a.y = lp[o1];

        v2f b0, b1, b2, b3;
        if (MASKED) {
            b0.x = rp0[o0] * bm0;  b0.y = rp0[o1] * bm0;
            b1.x = rp1[o0] * bm1;  b1.y = rp1[o1] * bm1;
            b2.x = rp2[o0] * bm2;  b2.y = rp2[o1] * bm2;
            b3.x = rp3[o0] * bm3;  b3.y = rp3[o1] * bm3;
        } else {
            b0.x = rp0[o0];  b0.y = rp0[o1];
            b1.x = rp1[o0];  b1.y = rp1[o1];
            b2.x = rp2[o0];  b2.y = rp2[o1];
            b3.x = rp3[o0];  b3.y = rp3[o1];
        }

        acc0 = __builtin_amdgcn_wmma_f32_16x16x4_f32(false, a, false, b0, (short)0, acc0, false, false);
        acc1 = __builtin_amdgcn_wmma_f32_16x16x4_f32(false, a, false, b1, (short)0, acc1, false, false);
        acc2 = __builtin_amdgcn_wmma_f32_16x16x4_f32(false, a, false, b2, (short)0, acc2, false, false);
        acc3 = __builtin_amdgcn_wmma_f32_16x16x4_f32(false, a, false, b3, (short)0, acc3, false, false);
    }
}

// Grid layout: blockIdx.x = (b, h/4, wt); the 4 waves of a block cover 4
// consecutive h at the same (b, wt). wt is a function of blockIdx.x only,
// so the boundary-mask specialization below is a *scalar* branch and every
// WMMA executes with EXEC all-1s (ISA requirement).
__global__ __launch_bounds__(128) void cost_volume_wmma_f32(
    const float* __restrict__ fl,   // [B,C,H,W]
    const float* __restrict__ fr,   // [B,C,H,W]
    float* __restrict__ out)        // [B,D,H,W]
{
    const int lane = threadIdx.x & 31;
    const int wave = threadIdx.x >> 5;

    const int wt = blockIdx.x & (WTILES - 1);          // uniform across block
    const int hq = (blockIdx.x >> 4) & 31;             // h / 4
    const int b  = blockIdx.x >> 9;                    // / (16*32)
    const int h  = hq * 4 + wave;                      // wave-uniform

    const int hf  = lane >> 4;                         // lane half: 0 or 1
    const int col = lane & 15;                         // M (for A) / N (for B,C,D)
    const int w0  = wt * 16;

    const size_t rowBase = ((size_t)b * CH) * HWPLANE + (size_t)h * WW;

    // A: lane's column of L^T -> element L[c][w0+col], c varies with K
    const float* lp = fl + rowBase + (size_t)(w0 + col);

    // B tiles j=0..3: columns w' = w0-48+16j+col ; clamp + 0/1 mask for w'<0
    const float* rp[4];
    float        bm[4];
    #pragma unroll
    for (int j = 0; j < 4; ++j) {
        const int wp = w0 - 48 + 16 * j + col;
        bm[j] = (wp >= 0) ? 1.0f : 0.0f;
        rp[j] = fr + rowBase + (size_t)((wp >= 0) ? wp : 0);
    }

    v8f acc0 = {}, acc1 = {}, acc2 = {}, acc3 = {};

    // w0 depends only on blockIdx.x -> scalar (s_cbranch) branch.
    if (w0 >= 48) {
        cv_kloop<false>(lp, rp[0], rp[1], rp[2], rp[3],
                        1.0f, 1.0f, 1.0f, 1.0f, hf, acc0, acc1, acc2, acc3);
    } else {
        cv_kloop<true>(lp, rp[0], rp[1], rp[2], rp[3],
                       bm[0], bm[1], bm[2], bm[3], hf, acc0, acc1, acc2, acc3);
    }

    // Scatter the band d = w - w' = 48 - 16j + m - n into out[b,d,h,w].
    // C/D layout: VGPR r, lane half hf, col n  ->  m = r + 8*hf, n = col.
    const float scale = 1.0f / (float)CH;
    v8f accs[4] = { acc0, acc1, acc2, acc3 };

    #pragma unroll
    for (int j = 0; j < 4; ++j) {
        #pragma unroll
        for (int r = 0; r < 8; ++r) {
            const int m = r + 8 * hf;
            const int d = 48 - 16 * j + m - col;
            if (d >= 0 && d < NDISP) {
                const int w = w0 + m;
                const size_t oidx = (((size_t)b * NDISP + d) * HH + h) * WW + w;
                out[oidx] = accs[j][r] * scale;
            }
        }
    }
}

extern "C" void kernel_launch(void* const* d_in, const int* in_sizes, int n_in,
                              void* d_out, int out_size, void* d_ws, size_t ws_size,
                              hipStream_t stream) {
    const float* feat_l = (const float*)d_in[0];
    const float* feat_r = (const float*)d_in[1];
    float* out = (float*)d_out;

    // B * (H/4) * WTILES blocks of 128 threads (4 waves); exact cover.
    const int blocks = BATCH * (HH / 4) * WTILES;      // 4096

    cost_volume_wmma_f32<<<blocks, 128, 0, stream>>>(feat_l, feat_r, out);
}